// EGAT_40200893890741
// MI455X (gfx1250) — compile-verified
//
#include <hip/hip_runtime.h>
#include <hip/hip_bf16.h>

// ---- problem constants ----
#define G_HEADS 4
#define G_HID   32
#define G_HC    128   // heads*hid
#define G_OUT   40
#define G_EDIM  5
#define NEG_BIG -3.402823e38f

typedef float v2f __attribute__((ext_vector_type(2)));
typedef float v8f __attribute__((ext_vector_type(8)));

// ---------------- utility ----------------
__device__ __forceinline__ void atomicMaxF(float* addr, float val) {
    if (val >= 0.f)
        atomicMax((int*)addr, __float_as_int(val));
    else
        atomicMin((unsigned int*)addr, __float_as_uint(val));
}

__global__ void fill_f32(float* __restrict__ p, int n, float v) {
    int i = blockIdx.x * blockDim.x + threadIdx.x;
    int stride = gridDim.x * blockDim.x;
    for (; i < n; i += stride) p[i] = v;
}

// degree count + self-loop attr accumulation
__global__ void deg_loopattr(const int* __restrict__ ei, const float* __restrict__ eattr,
                             float* __restrict__ cnt, float* __restrict__ lattr, int E) {
    int e = blockIdx.x * blockDim.x + threadIdx.x;
    if (e >= E) return;
    int d = ei[E + e];
    atomicAdd(&cnt[d], 1.f);
#pragma unroll
    for (int k = 0; k < G_EDIM; ++k)
        atomicAdd(&lattr[(long)d * G_EDIM + k], eattr[(long)e * G_EDIM + k]);
}

__global__ void loopattr_div(float* __restrict__ lattr, const float* __restrict__ cnt, int N) {
    int n = blockIdx.x * blockDim.x + threadIdx.x;
    if (n >= N) return;
    float c = fmaxf(cnt[n], 1.f);
#pragma unroll
    for (int k = 0; k < G_EDIM; ++k) lattr[(long)n * G_EDIM + k] /= c;
}

// fold We @ att_edge into tiny per-layer constant vectors:
// v1[d][h] = sum_c We1[d, h*32+c] * ae1[h,c]   (5x4)
// v2[d]    = sum_c We2[d, c]      * ae2[c]     (5)
__global__ void precompute_edge_vec(const float* __restrict__ We1, const float* __restrict__ ae1,
                                    const float* __restrict__ We2, const float* __restrict__ ae2,
                                    float* __restrict__ v1, float* __restrict__ v2) {
    int t = threadIdx.x;
    if (t < G_EDIM * G_HEADS) {
        int d = t / G_HEADS, h = t % G_HEADS;
        float s = 0.f;
        for (int c = 0; c < G_HID; ++c)
            s += We1[d * G_HC + h * G_HID + c] * ae1[h * G_HID + c];
        v1[d * G_HEADS + h] = s;
    } else if (t < G_EDIM * G_HEADS + G_EDIM) {
        int d = t - G_EDIM * G_HEADS;
        float s = 0.f;
        for (int c = 0; c < G_OUT; ++c)
            s += We2[d * G_OUT + c] * ae2[c];
        v2[d] = s;
    }
}

// ---------------- f32 WMMA GEMM: C[M,Ncols] = A[M,K] @ B[K,Ncols] ----------------
// One wave computes one 16x16 output tile via V_WMMA_F32_16X16X4_F32, K-loop step 4.
// Out-of-range output columns are CLAMPED (not branched): WMMA columns are
// independent, so dead columns compute garbage from in-bounds addresses and are
// simply never stored. Hot loop is branch-free with full EXEC (WMMA requirement).
__global__ void __launch_bounds__(256)
wmma_gemm_f32(const float* __restrict__ A, const float* __restrict__ B,
              float* __restrict__ C, int M, int K, int Ncols) {
    int wave = (int)((blockIdx.x * (long)blockDim.x + threadIdx.x) >> 5);
    int lane = threadIdx.x & 31;
    int tilesN = (Ncols + 15) >> 4;
    int bi = wave / tilesN;
    int bj = wave - bi * tilesN;
    if (bi * 16 >= M) return;
    int m0 = bi * 16, n0 = bj * 16;
    int mrow = m0 + (lane & 15);
    int ncol = n0 + (lane & 15);
    bool colok = (ncol < Ncols);
    int nc = colok ? ncol : (Ncols - 1);     // clamp: keep loads in-bounds
    int kh = (lane >> 4) << 1;               // 0 for lanes 0-15, 2 for lanes 16-31

    const float* Ap = A + (long)mrow * K + kh;          // A[m, kh]
    const float* Bp = B + (long)kh * Ncols + nc;        // B[kh, nc]
    const long  bstep = (long)4 * Ncols;

    v8f acc = {};
#pragma unroll 4
    for (int k = 0; k < K; k += 4) {
        v2f a = *(const v2f*)Ap;             // A[m, k+kh], A[m, k+kh+1] (contiguous b64)
        v2f b;
        b.x = Bp[0];                         // B[k+kh,   nc]
        b.y = Bp[Ncols];                     // B[k+kh+1, nc]
        acc = __builtin_amdgcn_wmma_f32_16x16x4_f32(
            false, a, false, b, (short)0, acc, false, false);
        Ap += 4;
        Bp += bstep;
    }
    // C/D layout: VGPR r -> M = r + (lane>>4)*8, N = lane&15
    int mb = m0 + ((lane >> 4) << 3);
    if (colok) {
#pragma unroll
        for (int r = 0; r < 8; ++r)
            C[(long)(mb + r) * Ncols + ncol] = acc[r];
    }
}

// ---------------- layer-1 attention ----------------
__global__ void alpha_sd1(const float* __restrict__ xs, const float* __restrict__ a_src,
                          const float* __restrict__ a_dst, float* __restrict__ as,
                          float* __restrict__ ad, int N) {
    int idx = blockIdx.x * blockDim.x + threadIdx.x;
    if (idx >= N * G_HEADS) return;
    int n = idx >> 2, h = idx & 3;
    const float* row = xs + (long)n * G_HC + h * G_HID;
    float s = 0.f, d = 0.f;
#pragma unroll
    for (int c = 0; c < G_HID; ++c) {
        float v = row[c];
        s += v * a_src[h * G_HID + c];
        d += v * a_dst[h * G_HID + c];
    }
    as[idx] = s; ad[idx] = d;
}

__global__ void edge_alpha_max1(const int* __restrict__ ei, const float* __restrict__ eattr,
                                const float* __restrict__ lattr, const float* __restrict__ v1,
                                const float* __restrict__ as, const float* __restrict__ ad,
                                float* __restrict__ raw, float* __restrict__ amax,
                                int E, int Etot) {
    int idx = blockIdx.x * blockDim.x + threadIdx.x;
    if (idx >= Etot * G_HEADS) return;
    int e = idx >> 2, h = idx & 3;
    int s, d; const float* ea;
    if (e < E) { s = ei[e]; d = ei[E + e]; ea = eattr + (long)e * G_EDIM; }
    else       { s = d = e - E;            ea = lattr + (long)(e - E) * G_EDIM; }
    float av = 0.f;
#pragma unroll
    for (int k = 0; k < G_EDIM; ++k) av += ea[k] * v1[k * G_HEADS + h];
    float a = as[s * G_HEADS + h] + ad[d * G_HEADS + h] + av;
    a = (a > 0.f) ? a : 0.2f * a;            // leaky_relu(0.2)
    raw[idx] = a;
    atomicMaxF(&amax[d * G_HEADS + h], a);
}

__global__ void edge_exp_sum1(const int* __restrict__ ei, float* __restrict__ raw,
                              const float* __restrict__ amax, float* __restrict__ den,
                              int E, int Etot) {
    int idx = blockIdx.x * blockDim.x + threadIdx.x;
    if (idx >= Etot * G_HEADS) return;
    int e = idx >> 2, h = idx & 3;
    int d = (e < E) ? ei[E + e] : (e - E);
    float ex = __expf(raw[idx] - amax[d * G_HEADS + h]);
    raw[idx] = ex;
    atomicAdd(&den[d * G_HEADS + h], ex);
}

// block-per-edge, 128 lanes = feature columns, coalesced gather + atomic scatter
__global__ void edge_aggregate1(const int* __restrict__ ei, const float* __restrict__ ex,
                                const float* __restrict__ den, const float* __restrict__ xs,
                                float* __restrict__ out, int E) {
    int e = blockIdx.x;
    int c = threadIdx.x;                    // 0..127
    __shared__ int ss, sd;
    __shared__ float coef[G_HEADS];
    if (c == 0) {
        if (e < E) { ss = ei[e]; sd = ei[E + e]; } else { ss = sd = e - E; }
    }
    __syncthreads();
    int s = ss, d = sd;
    if (c < G_HEADS)
        coef[c] = ex[(long)e * G_HEADS + c] / (den[d * G_HEADS + c] + 1e-16f);
    __syncthreads();
    float w = coef[c >> 5];
    atomicAdd(&out[(long)d * G_HC + c], xs[(long)s * G_HC + c] * w);
}

// out1 = elu(batchnorm(out1 + b1)) in place
__global__ void bn_elu1(float* __restrict__ h, const float* __restrict__ b,
                        const float* __restrict__ g, const float* __restrict__ be,
                        const float* __restrict__ rm, const float* __restrict__ rv, int N) {
    long idx = blockIdx.x * (long)blockDim.x + threadIdx.x;
    if (idx >= (long)N * G_HC) return;
    int c = (int)(idx & (G_HC - 1));
    float v = h[idx] + b[c];
    v = g[c] * (v - rm[c]) * rsqrtf(rv[c] + 1e-5f) + be[c];
    h[idx] = (v > 0.f) ? v : (__expf(v) - 1.f);
}

// ---------------- layer-2 attention (heads=1, C=40) ----------------
__global__ void alpha_sd2(const float* __restrict__ xs, const float* __restrict__ a_src,
                          const float* __restrict__ a_dst, float* __restrict__ as,
                          float* __restrict__ ad, int N) {
    int n = blockIdx.x * blockDim.x + threadIdx.x;
    if (n >= N) return;
    float s = 0.f, d = 0.f;
#pragma unroll
    for (int c = 0; c < G_OUT; ++c) {
        float v = xs[(long)n * G_OUT + c];
        s += v * a_src[c];
        d += v * a_dst[c];
    }
    as[n] = s; ad[n] = d;
}

__global__ void edge_alpha_max2(const int* __restrict__ ei, const float* __restrict__ eattr,
                                const float* __restrict__ lattr, const float* __restrict__ v2,
                                const float* __restrict__ as, const float* __restrict__ ad,
                                float* __restrict__ raw, float* __restrict__ amax,
                                int E, int Etot) {
    int e = blockIdx.x * blockDim.x + threadIdx.x;
    if (e >= Etot) return;
    int s, d; const float* ea;
    if (e < E) { s = ei[e]; d = ei[E + e]; ea = eattr + (long)e * G_EDIM; }
    else       { s = d = e - E;            ea = lattr + (long)(e - E) * G_EDIM; }
    float av = 0.f;
#pragma unroll
    for (int k = 0; k < G_EDIM; ++k) av += ea[k] * v2[k];
    float a = as[s] + ad[d] + av;
    a = (a > 0.f) ? a : 0.2f * a;
    raw[e] = a;
    atomicMaxF(&amax[d], a);
}

__global__ void edge_exp_sum2(const int* __restrict__ ei, float* __restrict__ raw,
                              const float* __restrict__ amax, float* __restrict__ den,
                              int E, int Etot) {
    int e = blockIdx.x * blockDim.x + threadIdx.x;
    if (e >= Etot) return;
    int d = (e < E) ? ei[E + e] : (e - E);
    float ex = __expf(raw[e] - amax[d]);
    raw[e] = ex;
    atomicAdd(&den[d], ex);
}

__global__ void edge_aggregate2(const int* __restrict__ ei, const float* __restrict__ ex,
                                const float* __restrict__ den, const float* __restrict__ xs,
                                float* __restrict__ out, int E) {
    int e = blockIdx.x;
    int c = threadIdx.x;                    // 0..63, guard at 40
    __shared__ int ss, sd;
    __shared__ float coef;
    if (c == 0) {
        if (e < E) { ss = ei[e]; sd = ei[E + e]; } else { ss = sd = e - E; }
    }
    __syncthreads();
    int s = ss, d = sd;
    if (c == 0) coef = ex[e] / (den[d] + 1e-16f);
    __syncthreads();
    if (c < G_OUT)
        atomicAdd(&out[(long)d * G_OUT + c], xs[(long)s * G_OUT + c] * coef);
}

// final: bn2 + log_softmax over 40 classes, one thread per node
__global__ void bn_logsoftmax(const float* __restrict__ h, const float* __restrict__ b,
                              const float* __restrict__ g, const float* __restrict__ be,
                              const float* __restrict__ rm, const float* __restrict__ rv,
                              float* __restrict__ out, int N) {
    int n = blockIdx.x * blockDim.x + threadIdx.x;
    if (n >= N) return;
    float v[G_OUT];
    float m = NEG_BIG;
#pragma unroll
    for (int c = 0; c < G_OUT; ++c) {
        float t = h[(long)n * G_OUT + c] + b[c];
        t = g[c] * (t - rm[c]) * rsqrtf(rv[c] + 1e-5f) + be[c];
        v[c] = t;
        m = fmaxf(m, t);
    }
    float s = 0.f;
#pragma unroll
    for (int c = 0; c < G_OUT; ++c) s += __expf(v[c] - m);
    float lse = __logf(s) + m;
#pragma unroll
    for (int c = 0; c < G_OUT; ++c) out[(long)n * G_OUT + c] = v[c] - lse;
}

// ---------------- host launcher ----------------
extern "C" void kernel_launch(void* const* d_in, const int* in_sizes, int n_in,
                              void* d_out, int out_size, void* d_ws, size_t ws_size,
                              hipStream_t stream) {
    const float* x      = (const float*)d_in[0];
    const int*   ei     = (const int*)  d_in[1];
    const float* eattr  = (const float*)d_in[2];
    const float* W1     = (const float*)d_in[3];
    const float* asrc1  = (const float*)d_in[4];
    const float* adst1  = (const float*)d_in[5];
    const float* aedge1 = (const float*)d_in[6];
    const float* We1    = (const float*)d_in[7];
    const float* b1     = (const float*)d_in[8];
    const float* W2     = (const float*)d_in[9];
    const float* asrc2  = (const float*)d_in[10];
    const float* adst2  = (const float*)d_in[11];
    const float* aedge2 = (const float*)d_in[12];
    const float* We2    = (const float*)d_in[13];
    const float* b2     = (const float*)d_in[14];
    const float* g1     = (const float*)d_in[15];
    const float* be1    = (const float*)d_in[16];
    const float* rm1    = (const float*)d_in[17];
    const float* rv1    = (const float*)d_in[18];
    const float* g2     = (const float*)d_in[19];
    const float* be2    = (const float*)d_in[20];
    const float* rm2    = (const float*)d_in[21];
    const float* rv2    = (const float*)d_in[22];
    float* out = (float*)d_out;

    const int N = in_sizes[0] / G_HC;        // 50000
    const int E = in_sizes[2] / G_EDIM;      // 800000
    const int Etot = E + N;                  // with self loops

    // workspace layout (floats)
    float* ws = (float*)d_ws;
    size_t o = 0;
    auto take = [&](size_t n) { float* p = ws + o; o += n; return p; };
    float* cnt   = take((size_t)N);
    float* lattr = take((size_t)N * G_EDIM);
    float* xs1   = take((size_t)N * G_HC);
    float* as1   = take((size_t)N * G_HEADS);
    float* ad1   = take((size_t)N * G_HEADS);
    float* amax1 = take((size_t)N * G_HEADS);
    float* den1  = take((size_t)N * G_HEADS);
    float* ex1   = take((size_t)Etot * G_HEADS);
    float* out1  = take((size_t)N * G_HC);   // becomes h1 after bn_elu1
    float* xs2   = take((size_t)N * G_OUT);
    float* as2   = take((size_t)N);
    float* ad2   = take((size_t)N);
    float* amax2 = take((size_t)N);
    float* den2  = take((size_t)N);
    float* ex2   = take((size_t)Etot);
    float* out2  = take((size_t)N * G_OUT);
    float* v1c   = take(32);
    float* v2c   = take(8);

    const int T = 256;
    auto blk = [](long n, int t) { return (int)((n + t - 1) / t); };

    // ---- init ----
    fill_f32<<<blk((long)N, T), T, 0, stream>>>(cnt, N, 0.f);
    fill_f32<<<blk((long)N * G_EDIM, T), T, 0, stream>>>(lattr, N * G_EDIM, 0.f);
    fill_f32<<<blk((long)N * G_HEADS, T), T, 0, stream>>>(amax1, N * G_HEADS, NEG_BIG);
    fill_f32<<<blk((long)N * G_HEADS, T), T, 0, stream>>>(den1, N * G_HEADS, 0.f);
    fill_f32<<<blk((long)N * G_HC, T), T, 0, stream>>>(out1, N * G_HC, 0.f);
    fill_f32<<<blk((long)N, T), T, 0, stream>>>(amax2, N, NEG_BIG);
    fill_f32<<<blk((long)N, T), T, 0, stream>>>(den2, N, 0.f);
    fill_f32<<<blk((long)N * G_OUT, T), T, 0, stream>>>(out2, N * G_OUT, 0.f);

    // ---- self-loop edge_attr = mean of incoming ----
    deg_loopattr<<<blk(E, T), T, 0, stream>>>(ei, eattr, cnt, lattr, E);
    loopattr_div<<<blk(N, T), T, 0, stream>>>(lattr, cnt, N);

    // ---- fold We @ att_edge into tiny constants ----
    precompute_edge_vec<<<1, 32, 0, stream>>>(We1, aedge1, We2, aedge2, v1c, v2c);

    // ---- layer 1 ----
    {
        int tilesN = G_HC / 16;                        // 8
        long waves = (long)((N + 15) / 16) * tilesN;   // 25000
        wmma_gemm_f32<<<blk(waves * 32, T), T, 0, stream>>>(x, W1, xs1, N, G_HC, G_HC);
    }
    alpha_sd1<<<blk((long)N * G_HEADS, T), T, 0, stream>>>(xs1, asrc1, adst1, as1, ad1, N);
    edge_alpha_max1<<<blk((long)Etot * G_HEADS, T), T, 0, stream>>>(
        ei, eattr, lattr, v1c, as1, ad1, ex1, amax1, E, Etot);
    edge_exp_sum1<<<blk((long)Etot * G_HEADS, T), T, 0, stream>>>(ei, ex1, amax1, den1, E, Etot);
    edge_aggregate1<<<Etot, 128, 0, stream>>>(ei, ex1, den1, xs1, out1, E);
    bn_elu1<<<blk((long)N * G_HC, T), T, 0, stream>>>(out1, b1, g1, be1, rm1, rv1, N);

    // ---- layer 2 ----
    {
        int tilesN = (G_OUT + 15) / 16;                // 3
        long waves = (long)((N + 15) / 16) * tilesN;   // 9375
        wmma_gemm_f32<<<blk(waves * 32, T), T, 0, stream>>>(out1, W2, xs2, N, G_HC, G_OUT);
    }
    alpha_sd2<<<blk(N, T), T, 0, stream>>>(xs2, asrc2, adst2, as2, ad2, N);
    edge_alpha_max2<<<blk(Etot, T), T, 0, stream>>>(
        ei, eattr, lattr, v2c, as2, ad2, ex2, amax2, E, Etot);
    edge_exp_sum2<<<blk(Etot, T), T, 0, stream>>>(ei, ex2, amax2, den2, E, Etot);
    edge_aggregate2<<<Etot, 64, 0, stream>>>(ei, ex2, den2, xs2, out2, E);

    // ---- bn2 + log_softmax ----
    bn_logsoftmax<<<blk(N, T), T, 0, stream>>>(out2, b2, g2, be2, rm2, rv2, out, N);
}